// HouseholderRotation_78735340470884
// MI455X (gfx1250) — compile-verified
//
#include <hip/hip_runtime.h>
#include <hip/hip_bf16.h>

typedef float v2f __attribute__((ext_vector_type(2)));
typedef float v8f __attribute__((ext_vector_type(8)));

#define NREF 16
#define NF   4096
#define EPSF 1e-6f
#define NWAVE 4
#define KSLICE (NF / NWAVE)   // 1024

// ---------------------------------------------------------------------------
// Kernel 1 (one workgroup): build W' = -(T^T * V) where T is the 16x16
// lower-triangular WY factor of the Householder product H16...H1 = I - V T V^T.
// ---------------------------------------------------------------------------
__global__ __launch_bounds__(256) void hh_build_w(const float* __restrict__ vec,
                                                  float* __restrict__ wp) {
    __shared__ float G[NREF][NREF];
    __shared__ float T[NREF][NREF];
    const int t = threadIdx.x;       // 256 threads
    const int i = t >> 4;
    const int j = t & 15;

    // Gram matrix G[i][j] = v_i . v_j  (one (i,j) pair per thread)
    {
        const float* vi = vec + (size_t)i * NF;
        const float* vj = vec + (size_t)j * NF;
        float acc = 0.f;
        for (int k = 0; k < NF; ++k) acc = fmaf(vi[k], vj[k], acc);
        G[i][j] = acc;
    }
    __syncthreads();

    // Serial (tiny) triangular recurrence for T
    if (t == 0) {
        for (int kk = 0; kk < NREF; ++kk) {
            const float beta = 2.0f / (G[kk][kk] + EPSF);
            for (int jj = 0; jj < kk; ++jj) {
                float s = 0.f;
                for (int ii = jj; ii < kk; ++ii) s = fmaf(G[kk][ii], T[ii][jj], s);
                T[kk][jj] = -beta * s;
            }
            T[kk][kk] = beta;
            for (int jj = kk + 1; jj < NREF; ++jj) T[kk][jj] = 0.f;
        }
    }
    __syncthreads();

    // W'[i][k] = -sum_j T[j][i] * vec[j][k]; thread handles row i, 256-col slab
    float trow[NREF];
#pragma unroll
    for (int jj = 0; jj < NREF; ++jj) trow[jj] = T[jj][i];
    const int k0 = j * (NF / NREF);
    for (int k = k0; k < k0 + NF / NREF; ++k) {
        float s = 0.f;
#pragma unroll
        for (int jj = 0; jj < NREF; ++jj) s = fmaf(trow[jj], vec[(size_t)jj * NF + k], s);
        wp[(size_t)i * NF + k] = -s;
    }
}

// ---------------------------------------------------------------------------
// Kernel 2: one workgroup (4 waves) per 16-row tile; 2048 waves total.
//   Phase A: wave w accumulates K-slice [w*1024,(w+1)*1024) of
//            P(16x16) = X_tile(16x4096) * V^T   via v_wmma_f32_16x16x4_f32,
//            partials reduced through LDS.
//   Phase B: wave w emits column slab [w*1024,(w+1)*1024) of
//            out_tile = X_tile + P * W'          via v_wmma_f32_16x16x4_f32.
// ---------------------------------------------------------------------------
__global__ __launch_bounds__(NWAVE * 32) void hh_apply(const float* __restrict__ x,
                                                       const float* __restrict__ vec,
                                                       const float* __restrict__ wp,
                                                       float* __restrict__ out) {
    __shared__ float pl[NWAVE][NREF][NREF + 1];  // pitch 17: conflict-free
    const int w    = threadIdx.x >> 5;           // wave id 0..3
    const int lane = threadIdx.x & 31;           // wave32
    const int half = lane >> 4;                  // 0: K=+0,+1  1: K=+2,+3 (f32 frag layout)
    const int m    = lane & 15;
    const size_t row0 = (size_t)blockIdx.x * NREF;

    // ---- Phase A: partial P over K-slice, steps of 4 ----
    const int kbase = w * KSLICE;
    const float* xa = x   + (row0 + (size_t)m) * NF + kbase + 2 * half; // A[m][k..k+1]
    const float* vb = vec + (size_t)m * NF + kbase + 2 * half;          // B[k][j]=vec[j][k]
    v8f p = {0.f, 0.f, 0.f, 0.f, 0.f, 0.f, 0.f, 0.f};
#pragma unroll 8
    for (int k = 0; k < KSLICE; k += 4) {
        v2f a = *(const v2f*)(xa + k);
        v2f b = *(const v2f*)(vb + k);
        p = __builtin_amdgcn_wmma_f32_16x16x4_f32(false, a, false, b,
                                                  (short)0, p, false, false);
    }

    // D-layout -> LDS: VGPR r holds (M = r + 8*half, N = m)
#pragma unroll
    for (int r = 0; r < 8; ++r) pl[w][r + 8 * half][m] = p[r];
    __syncthreads();

    // Gather P as four 16x4 A-fragments, reducing the 4 K-slice partials
    v2f af[4];
#pragma unroll
    for (int tt = 0; tt < 4; ++tt) {
        float ax = 0.f, ay = 0.f;
#pragma unroll
        for (int ww = 0; ww < NWAVE; ++ww) {
            ax += pl[ww][m][4 * tt + 2 * half];
            ay += pl[ww][m][4 * tt + 2 * half + 1];
        }
        af[tt].x = ax;
        af[tt].y = ay;
    }

    // ---- Phase B: out = X + P * W' over this wave's 1024-column slab ----
    const int nbase = w * KSLICE;
    const float* xc = x   + row0 * NF + m;
    float*       oc = out + row0 * NF + m;
    for (int n0 = nbase; n0 < nbase + KSLICE; n0 += 16) {
        v8f c;
#pragma unroll
        for (int r = 0; r < 8; ++r) c[r] = xc[(size_t)(r + 8 * half) * NF + n0];
#pragma unroll
        for (int tt = 0; tt < 4; ++tt) {
            const float* wb = wp + (size_t)(4 * tt + 2 * half) * NF + n0 + m;
            v2f b;
            b.x = wb[0];
            b.y = wb[NF];
            c = __builtin_amdgcn_wmma_f32_16x16x4_f32(false, af[tt], false, b,
                                                      (short)0, c, false, false);
        }
#pragma unroll
        for (int r = 0; r < 8; ++r) oc[(size_t)(r + 8 * half) * NF + n0] = c[r];
    }
}

// ---------------------------------------------------------------------------
extern "C" void kernel_launch(void* const* d_in, const int* in_sizes, int n_in,
                              void* d_out, int out_size, void* d_ws, size_t ws_size,
                              hipStream_t stream) {
    const float* x   = (const float*)d_in[0];   // (4,2048,4096) f32
    const float* vec = (const float*)d_in[1];   // (16,4096) f32
    float*       out = (float*)d_out;           // f32
    float*       wp  = (float*)d_ws;            // needs 16*4096*4 = 256 KB

    hh_build_w<<<1, 256, 0, stream>>>(vec, wp);

    const int rows = in_sizes[0] / NF;          // 8192
    hh_apply<<<rows / NREF, NWAVE * 32, 0, stream>>>(x, vec, wp, out);
}